// SparseBasicBlock3d_81793357185328
// MI455X (gfx1250) — compile-verified
//
#include <hip/hip_runtime.h>
#include <hip/hip_bf16.h>
#include <math.h>

typedef __attribute__((ext_vector_type(2))) float v2f;
typedef __attribute__((ext_vector_type(4))) float v4f;
typedef __attribute__((ext_vector_type(8))) float v8f;

#define GD 128
#define GRID_VOX (GD * GD * GD)   // 2097152
#define NROWS 64                  // rows (active sites) per block tile
#define LSTR 68                   // LDS row stride in floats (pad: 16B aligned + conflict-free)

// ---------------- hash grid (rulebook) ----------------
__global__ void grid_init_kernel(int* __restrict__ grid) {
    int i = blockIdx.x * 256 + threadIdx.x;
    if (i < GRID_VOX) grid[i] = -1;
}

__global__ void grid_build_kernel(const int* __restrict__ coords,
                                  int* __restrict__ grid, int N) {
    int i = blockIdx.x * 256 + threadIdx.x;
    if (i < N) {
        int z = coords[3 * i + 0];
        int y = coords[3 * i + 1];
        int x = coords[3 * i + 2];
        grid[(z * GD + y) * GD + x] = i;
    }
}

// ---------------- weight packing into WMMA B-fragment order ----------------
// B-matrix 4x16 f32 layout for v_wmma_f32_16x16x4_f32 (wave32):
//   lane L in [0,16):  holds column N=L,    {K=0, K=1} in the v2f
//   lane L in [16,32): holds column N=L-16, {K=2, K=3}
// packed[((off*4 + ntile)*16 + kstep)*32 + lane] = float2{ w[4k+2h][col], w[4k+2h+1][col] }
__global__ void pack_w1_kernel(const float* __restrict__ w1, float* __restrict__ w1p) {
    int t = blockIdx.x * 256 + threadIdx.x;
    if (t >= 27 * 4 * 16 * 32) return;
    int lane = t & 31;
    int k    = (t >> 5) & 15;
    int nt   = (t >> 9) & 3;
    int off  = t >> 11;
    int half = lane >> 4;
    int col  = nt * 16 + (lane & 15);
    int krow = 4 * k + 2 * half;
    const float* w = w1 + (size_t)off * 64 * 64;   // [Cin=64][Cout=64]
    w1p[(size_t)t * 2 + 0] = w[(krow + 0) * 64 + col];
    w1p[(size_t)t * 2 + 1] = w[(krow + 1) * 64 + col];
}

__global__ void pack_w2_kernel(const float* __restrict__ w2, float* __restrict__ w2p) {
    int t = blockIdx.x * 256 + threadIdx.x;
    if (t >= 4 * 16 * 32) return;
    int lane = t & 31;
    int k    = (t >> 5) & 15;
    int nt   = (t >> 9) & 3;
    int half = lane >> 4;
    int col  = nt * 16 + (lane & 15);
    int krow = 4 * k + 2 * half;
    w2p[(size_t)t * 2 + 0] = w2[(krow + 0) * 64 + col];
    w2p[(size_t)t * 2 + 1] = w2[(krow + 1) * 64 + col];
}

// ---------------- fused conv3d(27 taps) + BN + ReLU + 1x1 conv ----------------
// Block: 128 threads = 4 wave32; tile: 64 rows x 64 Cout.
// Wave w owns Cout columns [16w, 16w+16).
__global__ __launch_bounds__(128)
void sparse_block_kernel(const float* __restrict__ feat,
                         const int*   __restrict__ coords,
                         const int*   __restrict__ grid,
                         const float* __restrict__ w1p,
                         const float* __restrict__ b1,
                         const float* __restrict__ gamma,
                         const float* __restrict__ beta,
                         const float* __restrict__ rmean,
                         const float* __restrict__ rvar,
                         const float* __restrict__ w2p,
                         const float* __restrict__ b2,
                         float* __restrict__ out,
                         int N) {
    __shared__ float lds[NROWS * LSTR];   // gathered A tile, then reused for activation Y
    __shared__ int   snidx[NROWS];
    __shared__ int   scz[NROWS], scy[NROWS], scx[NROWS];

    const int tid  = threadIdx.x;
    const int lane = tid & 31;
    const int wv   = tid >> 5;          // 0..3 : Cout ntile
    const int half = lane >> 4;         // 0/1  : which half of the wave
    const int l15  = lane & 15;
    const int col  = wv * 16 + l15;     // output channel owned by this lane
    const int row0 = blockIdx.x * NROWS;

    // cache this tile's coordinates
    if (tid < NROWS) {
        int r = row0 + tid;
        if (r < N) {
            scz[tid] = coords[3 * r + 0];
            scy[tid] = coords[3 * r + 1];
            scx[tid] = coords[3 * r + 2];
        } else {
            scz[tid] = -10000; scy[tid] = -10000; scx[tid] = -10000;
        }
    }
    __syncthreads();

    v8f acc[4];
    #pragma unroll
    for (int m = 0; m < 4; ++m) acc[m] = (v8f)0.0f;

    for (int off = 0; off < 27; ++off) {
        const int dz = off / 9 - 1;
        const int dy = (off / 3) % 3 - 1;
        const int dx = off % 3 - 1;

        // rulebook lookup for this tap
        if (tid < NROWS) {
            int nz = scz[tid] + dz, ny = scy[tid] + dy, nx = scx[tid] + dx;
            int idx = -1;
            if (nz >= 0 && nz < GD && ny >= 0 && ny < GD && nx >= 0 && nx < GD)
                idx = grid[(nz * GD + ny) * GD + nx];
            snidx[tid] = idx;
        }
        __syncthreads();

        // gather 64 rows x 64 f32 into LDS (zero rows for inactive neighbors)
        #pragma unroll
        for (int it = 0; it < (NROWS * 16) / 128; ++it) {
            int i = it * 128 + tid;        // row*16 + q (q = which float4)
            int r = i >> 4;
            int q = i & 15;
            int src = snidx[r];
            v4f v = (v4f)0.0f;
            if (src >= 0)
                v = *(const v4f*)(feat + (size_t)src * 64 + q * 4);
            *(v4f*)(&lds[r * LSTR + q * 4]) = v;
        }
        __syncthreads();

        // GEMM: acc[m] += A(16x4) x B(4x16) over 16 K-steps, 4 M-subtiles
        const float* bp = w1p + (size_t)(off * 4 + wv) * 16 * 64;
        #pragma unroll
        for (int k = 0; k < 16; ++k) {
            v2f bf = *(const v2f*)(bp + k * 64 + lane * 2);
            #pragma unroll
            for (int m = 0; m < 4; ++m) {
                v2f af = *(const v2f*)(&lds[(m * 16 + l15) * LSTR + 4 * k + 2 * half]);
                acc[m] = __builtin_amdgcn_wmma_f32_16x16x4_f32(
                    false, af, false, bf, (short)0, acc[m], false, false);
            }
        }
        __syncthreads();   // protect LDS before next tap's gather
    }

    // bias + BN(eval) + ReLU, fold constants per output channel (constant per lane)
    const float scl = gamma[col] * rsqrtf(rvar[col] + 1e-5f);
    const float shf = (b1[col] - rmean[col]) * scl + beta[col];

    // write activation Y into LDS (C/D layout: VGPR j -> row m*16 + j + 8*half, col = lane col)
    #pragma unroll
    for (int m = 0; m < 4; ++m) {
        #pragma unroll
        for (int j = 0; j < 8; ++j) {
            float y = acc[m][j] * scl + shf;
            y = y > 0.0f ? y : 0.0f;
            lds[(m * 16 + j + half * 8) * LSTR + col] = y;
        }
    }
    __syncthreads();

    // 1x1 conv: out_tile = Y(64x64) @ w2, same WMMA structure
    v8f acc2[4];
    #pragma unroll
    for (int m = 0; m < 4; ++m) acc2[m] = (v8f)0.0f;
    #pragma unroll
    for (int k = 0; k < 16; ++k) {
        v2f bf = *(const v2f*)(w2p + (size_t)(wv * 16 + k) * 64 + lane * 2);
        #pragma unroll
        for (int m = 0; m < 4; ++m) {
            v2f af = *(const v2f*)(&lds[(m * 16 + l15) * LSTR + 4 * k + 2 * half]);
            acc2[m] = __builtin_amdgcn_wmma_f32_16x16x4_f32(
                false, af, false, bf, (short)0, acc2[m], false, false);
        }
    }

    const float bb = b2[col];
    #pragma unroll
    for (int m = 0; m < 4; ++m) {
        int rg = row0 + m * 16 + half * 8;
        #pragma unroll
        for (int j = 0; j < 8; ++j) {
            int r = rg + j;
            if (r < N)
                out[(size_t)r * 64 + col] = acc2[m][j] + bb;
        }
    }
}

extern "C" void kernel_launch(void* const* d_in, const int* in_sizes, int n_in,
                              void* d_out, int out_size, void* d_ws, size_t ws_size,
                              hipStream_t stream) {
    const float* feat  = (const float*)d_in[0];
    const int*   coords= (const int*)  d_in[1];
    const float* w1    = (const float*)d_in[2];
    const float* b1    = (const float*)d_in[3];
    const float* gamma = (const float*)d_in[4];
    const float* beta  = (const float*)d_in[5];
    const float* rmean = (const float*)d_in[6];
    const float* rvar  = (const float*)d_in[7];
    const float* w2    = (const float*)d_in[8];
    const float* b2    = (const float*)d_in[9];
    float* out = (float*)d_out;

    const int N = in_sizes[0] / 64;

    char* ws = (char*)d_ws;
    int*   grid = (int*)ws;                                   // 8 MB
    float* w1p  = (float*)(ws + (size_t)GRID_VOX * 4);        // 27*4*16*32*2 f32 = 432 KB
    float* w2p  = w1p + (size_t)27 * 4 * 16 * 32 * 2;         // 16 KB

    grid_init_kernel<<<GRID_VOX / 256, 256, 0, stream>>>(grid);
    grid_build_kernel<<<(N + 255) / 256, 256, 0, stream>>>(coords, grid, N);
    pack_w1_kernel<<<(27 * 4 * 16 * 32 + 255) / 256, 256, 0, stream>>>(w1, w1p);
    pack_w2_kernel<<<(4 * 16 * 32 + 255) / 256, 256, 0, stream>>>(w2, w2p);

    const int nblk = (N + NROWS - 1) / NROWS;   // 400000/64 = 6250 exactly
    sparse_block_kernel<<<nblk, 128, 0, stream>>>(feat, coords, grid, w1p, b1,
                                                  gamma, beta, rmean, rvar,
                                                  w2p, b2, out, N);
}